// magnet_loss_80539226735047
// MI455X (gfx1250) — compile-verified
//
#include <hip/hip_runtime.h>
#include <math.h>

// ---------------------------------------------------------------------------
// Magnet loss on gfx1250 (MI455X).
// Heavy op: outputs[8192x2048] @ means.T[2048x256] done with
// V_WMMA_F32_16X16X4_F32 (fp32 matrix pipe, matches reference precision).
// Everything else is bandwidth-trivial (input fits in 192MB L2).
// ---------------------------------------------------------------------------

#define NROW 8192
#define DIMS 2048
#define NCLU 256
#define ALPHA_C 7.18f

typedef __attribute__((ext_vector_type(2))) float v2f;
typedef __attribute__((ext_vector_type(8))) float v8f;

// workspace layout (float offsets)
#define OFF_SUMS  0                       // [NCLU*DIMS] sums -> means (in place)
#define OFF_CNT   (NCLU * DIMS)           // [NCLU] counts
#define OFF_MN2   (OFF_CNT + NCLU)        // [NCLU] ||mean||^2
#define OFF_VAL   (OFF_MN2 + NCLU)        // [NCLU] valid flag (0/1)
#define OFF_XN2   (OFF_VAL + NCLU)        // [NROW] ||x||^2
#define OFF_RN    (OFF_XN2 + NROW)        // [NROW] residual norm
#define OFF_SCAL  (OFF_RN + NROW)         // [0]=sum_norm [1]=num_inst [2]=stdev [3]=loss
#define WS_TOTAL  (OFF_SCAL + 8)          // ~2.17 MB of float

__global__ void k_init(float* ws) {
    int i = blockIdx.x * blockDim.x + threadIdx.x;
    if (i < WS_TOTAL) ws[i] = 0.0f;
}

// one block per row: atomic segment-sum into sums/counts (2MB table, L2-resident)
__global__ void k_accum(const float* __restrict__ x, const int* __restrict__ clu,
                        float* ws) {
    int row = blockIdx.x;
    int c = clu[row];
    float* sums = ws + OFF_SUMS + (size_t)c * DIMS;
    const float* xr = x + (size_t)row * DIMS;
    for (int d = threadIdx.x; d < DIMS; d += blockDim.x)
        atomicAdd(&sums[d], xr[d]);
    if (threadIdx.x == 0) atomicAdd(&ws[OFF_CNT + c], 1.0f);
}

// one block per cluster: means in place, ||mean||^2, valid, num_instances
__global__ void k_means(float* ws) {
    __shared__ float red[256];
    int c = blockIdx.x;
    float cnt = ws[OFF_CNT + c];
    float inv = 1.0f / fmaxf(cnt, 1.0f);
    float* m = ws + OFF_SUMS + (size_t)c * DIMS;
    float acc = 0.0f;
    for (int d = threadIdx.x; d < DIMS; d += blockDim.x) {
        float v = m[d] * inv;
        m[d] = v;
        acc += v * v;
    }
    red[threadIdx.x] = acc;
    __syncthreads();
    for (int s = blockDim.x >> 1; s > 0; s >>= 1) {
        if (threadIdx.x < s) red[threadIdx.x] += red[threadIdx.x + s];
        __syncthreads();
    }
    if (threadIdx.x == 0) {
        ws[OFF_MN2 + c] = red[0];
        float v = (cnt >= 4.0f) ? 1.0f : 0.0f;
        ws[OFF_VAL + c] = v;
        if (v != 0.0f) atomicAdd(&ws[OFF_SCAL + 1], cnt);
    }
}

// one block per row: residual norm to own mean, ||x||^2, masked sum of norms
__global__ void k_resid(const float* __restrict__ x, const int* __restrict__ clu,
                        float* ws) {
    __shared__ float r1[256];
    __shared__ float r2[256];
    int row = blockIdx.x;
    int c = clu[row];
    const float* xr = x + (size_t)row * DIMS;
    const float* m = ws + OFF_SUMS + (size_t)c * DIMS;
    float a1 = 0.0f, a2 = 0.0f;
    for (int d = threadIdx.x; d < DIMS; d += blockDim.x) {
        float xv = xr[d];
        float dv = xv - m[d];
        a1 += dv * dv;
        a2 += xv * xv;
    }
    r1[threadIdx.x] = a1;
    r2[threadIdx.x] = a2;
    __syncthreads();
    for (int s = blockDim.x >> 1; s > 0; s >>= 1) {
        if (threadIdx.x < s) {
            r1[threadIdx.x] += r1[threadIdx.x + s];
            r2[threadIdx.x] += r2[threadIdx.x + s];
        }
        __syncthreads();
    }
    if (threadIdx.x == 0) {
        float rn = sqrtf(r1[0]);
        ws[OFF_RN + row] = rn;
        ws[OFF_XN2 + row] = r2[0];
        if (ws[OFF_VAL + c] != 0.0f) atomicAdd(&ws[OFF_SCAL + 0], rn);
    }
}

__global__ void k_stdev(float* ws) {
    float s = ws[OFF_SCAL + 0];
    float ni = ws[OFF_SCAL + 1];
    ws[OFF_SCAL + 2] = s * s / ni;
}

// ---------------------------------------------------------------------------
// GEMM + epilogue. 8 waves/block, each wave: 16-row tile x all 256 clusters.
// A (16x4 fp32): lane 0-15 = M rows, VGPR0 = K{0 | 2}, VGPR1 = K{1 | 3}.
// B (4x16 fp32): lane lo = N col,    VGPR0 = K{0 | 2}, VGPR1 = K{1 | 3}.
// C/D (16x16 fp32): VGPR j of lane L = row (j + 8*(L>>4)), col (L&15).
// ---------------------------------------------------------------------------
__global__ void __launch_bounds__(256) k_gemm(const float* __restrict__ x,
                                              const int* __restrict__ clu,
                                              float* ws) {
    const int lane = threadIdx.x & 31;
    const int wave = threadIdx.x >> 5;
    const int lo = lane & 15;
    const int hi = lane >> 4;
    const int m0 = blockIdx.x * 128 + wave * 16;

    const float* __restrict__ means = ws + OFF_SUMS;

    v8f acc[16] = {};

    const float* __restrict__ arow = x + (size_t)(m0 + lo) * DIMS + 2 * hi;
    const float* __restrict__ bbase = means + (size_t)lo * DIMS + 2 * hi;

    for (int k = 0; k < DIMS; k += 4) {
        v2f a;
        a.x = arow[k];
        a.y = arow[k + 1];
#pragma unroll
        for (int nt = 0; nt < 16; ++nt) {
            const float* bp = bbase + (size_t)(nt * 16) * DIMS + k;
            v2f b;
            b.x = bp[0];
            b.y = bp[1];
            acc[nt] = __builtin_amdgcn_wmma_f32_16x16x4_f32(
                false, a, false, b, (short)0, acc[nt], false, false);
        }
    }

    // epilogue: dist -> exp -> masked denom -> per-row loss
    const float stdev = ws[OFF_SCAL + 2];
    const float* __restrict__ mn2 = ws + OFF_MN2;
    const float* __restrict__ val = ws + OFF_VAL;

#pragma unroll
    for (int j = 0; j < 8; ++j) {
        int row = m0 + j + 8 * hi;
        float xn = ws[OFF_XN2 + row];
        int myc = clu[row];
        float partial = 0.0f;
#pragma unroll
        for (int nt = 0; nt < 16; ++nt) {
            int c = nt * 16 + lo;
            float dot = acc[nt][j];
            float sq = xn - 2.0f * dot + mn2[c];
            float dist = sqrtf(fmaxf(sq, 0.0f));
            float e = expf(-0.5f * dist / stdev);
            if ((val[c] != 0.0f) && (c != myc)) partial += e;
        }
        // reduce denom across the 16 lanes that share this row
#pragma unroll
        for (int off = 8; off > 0; off >>= 1)
            partial += __shfl_xor(partial, off, 16);
        if (lo == 0) {
            if (val[myc] != 0.0f) {
                float li = logf(partial) + 0.5f * ws[OFF_RN + row] / stdev + ALPHA_C;
                atomicAdd(&ws[OFF_SCAL + 3], li);
            }
        }
    }
}

__global__ void k_final(const float* __restrict__ ws, float* __restrict__ out) {
    out[0] = ws[OFF_SCAL + 3] / ws[OFF_SCAL + 1];
}

extern "C" void kernel_launch(void* const* d_in, const int* in_sizes, int n_in,
                              void* d_out, int out_size, void* d_ws, size_t ws_size,
                              hipStream_t stream) {
    const float* x = (const float*)d_in[0];   // [8192, 2048] fp32
    const int* clu = (const int*)d_in[1];     // [8192] int32
    float* ws = (float*)d_ws;                 // needs ~2.2 MB
    float* out = (float*)d_out;               // [1] fp32

    k_init<<<(WS_TOTAL + 255) / 256, 256, 0, stream>>>(ws);
    k_accum<<<NROW, 256, 0, stream>>>(x, clu, ws);
    k_means<<<NCLU, 256, 0, stream>>>(ws);
    k_resid<<<NROW, 256, 0, stream>>>(x, clu, ws);
    k_stdev<<<1, 1, 0, stream>>>(ws);
    k_gemm<<<NROW / 128, 256, 0, stream>>>(x, clu, ws);
    k_final<<<1, 1, 0, stream>>>(ws, out);
}